// IRLSEightPointNet_2783138808087
// MI455X (gfx1250) — compile-verified
//
#include <hip/hip_runtime.h>
#include <math.h>

// ---------------------------------------------------------------------------
// IRLS eight-point: fused-pass design for MI455X (gfx1250, wave32)
//   Pass A (1 stream of 117MB): centroid sums + raw 9x9 Kronecker moments M0
//           via V_WMMA_F32_16X16X4_F32 (fp32 matrix pipe, K=4 per op)
//   Pass B (L2-resident re-read): weighted mean radius (needs centroids)
//   Finalize: per-batch 9x9 Jacobi eigh + 3x3 rank-2 projection (tiny)
// ---------------------------------------------------------------------------

typedef float v2f __attribute__((ext_vector_type(2)));
typedef float v8f __attribute__((ext_vector_type(8)));

#define AUG 1e-4f
#define WAVES_PER_WG 8
#define TILE 32
#define TILES_PER_WAVE 4
#define CHUNK_PTS (WAVES_PER_WG * TILE * TILES_PER_WAVE) /* 1024 */

__global__ void zero_ws_kernel(float* __restrict__ p, int n) {
  int i = blockIdx.x * blockDim.x + threadIdx.x;
  if (i < n) p[i] = 0.0f;
}

// ---------------------------------------------------------------------------
// Pass A: per chunk of 1024 points of one batch.
//   sums[b*8 + {0..4}] += { w, w*x1, w*y1, w*x2, w*y2 }
//   M0[b*256 + m*16+n] += sum_k w_k^2 * xvec_k[m] * xvec_k[n]   (m,n < 9 used)
// Staged design row per point (scaled by w once; A*B contributes w^2):
//   xvec = [x1,y1,1] (outer) [w*x2, w*y2, w], padded to 16 with zeros.
// WMMA f32 16x16x4: A[m][k]=staged[k][m], B[k][n]=staged[k][n]; 4 points/op.
// ---------------------------------------------------------------------------
__global__ __launch_bounds__(256) void moments_kernel(
    const float* __restrict__ pts1, const float* __restrict__ pts2,
    const float* __restrict__ wgt, float* __restrict__ sums,
    float* __restrict__ M0, int B, int N, int chunks)
{
  __shared__ float stage[WAVES_PER_WG * TILE * 16]; // 16 KB, per-wave regions
  __shared__ float red[WAVES_PER_WG * 256];         // 8 KB, cross-wave reduce

  const int wg   = blockIdx.x;
  const int b    = wg / chunks;
  const int c    = wg % chunks;
  const int tid  = threadIdx.x;
  const int lane = tid & 31;
  const int wave = tid >> 5;
  const int m    = lane & 15;   // matrix row/col index this lane serves
  const int half = lane >> 4;   // 0: lanes 0-15, 1: lanes 16-31

  const float* p1b = pts1 + (size_t)b * N * 3;
  const float* p2b = pts2 + (size_t)b * N * 3;
  const float* wb  = wgt  + (size_t)b * N;

  v8f acc = {0.f, 0.f, 0.f, 0.f, 0.f, 0.f, 0.f, 0.f};
  float sw = 0.f, swx1 = 0.f, swy1 = 0.f, swx2 = 0.f, swy2 = 0.f;
  float* mystage = stage + wave * (TILE * 16);

  for (int t = 0; t < TILES_PER_WAVE; ++t) {
    int n = c * CHUNK_PTS + (wave * TILES_PER_WAVE + t) * TILE + lane;
    float x1 = 0.f, y1 = 0.f, x2 = 0.f, y2 = 0.f, wv = 0.f;
    if (n < N) {
      x1 = p1b[3 * n + 0]; y1 = p1b[3 * n + 1];
      x2 = p2b[3 * n + 0]; y2 = p2b[3 * n + 1];
      wv = wb[n];
    }
    sw += wv; swx1 += wv * x1; swy1 += wv * y1; swx2 += wv * x2; swy2 += wv * y2;

    // stage 16-float design row for this lane's point (64B aligned -> b128 ds)
    float q0 = wv * x2, q1 = wv * y2, q2 = wv;
    float4* r4 = (float4*)(mystage + lane * 16);
    r4[0] = make_float4(x1 * q0, x1 * q1, x1 * q2, y1 * q0);
    r4[1] = make_float4(y1 * q1, y1 * q2, q0, q1);
    r4[2] = make_float4(q2, 0.f, 0.f, 0.f);
    r4[3] = make_float4(0.f, 0.f, 0.f, 0.f);
    __syncthreads();

#pragma unroll
    for (int g = 0; g < 8; ++g) {            // 8 groups x 4 points = 32 points
      const float* gp = mystage + g * 4 * 16;
      v2f Am, Bm;
      // A 16x4 layout: VGPR v holds K = v + 2*half  (ISA 7.12.2)
      Am.x = gp[(2 * half + 0) * 16 + m];
      Am.y = gp[(2 * half + 1) * 16 + m];
      // B 4x16 layout: VGPR v holds row K = v*2 + half, col = lane%16
      Bm.x = gp[(half + 0) * 16 + m];
      Bm.y = gp[(half + 2) * 16 + m];
      acc = __builtin_amdgcn_wmma_f32_16x16x4_f32(
          /*neg_a=*/false, Am, /*neg_b=*/false, Bm,
          /*c_mod=*/(short)0, acc, /*reuse_a=*/false, /*reuse_b=*/false);
    }
    __syncthreads();
  }

  // Cross-wave reduction of 16x16 accumulators through LDS.
  // C/D layout: VGPR v, lane -> D[v + 8*half][lane%16]
  {
    float* rr = red + wave * 256;
#pragma unroll
    for (int v = 0; v < 8; ++v) rr[(v + 8 * half) * 16 + m] = acc[v];
  }
  __syncthreads();
  {
    float s = 0.f;
#pragma unroll
    for (int w2 = 0; w2 < WAVES_PER_WG; ++w2) s += red[w2 * 256 + tid];
    atomicAdd(&M0[(size_t)b * 256 + tid], s);
  }

  // wave32 shuffle reduction of centroid partial sums
  for (int o = 16; o > 0; o >>= 1) {
    sw   += __shfl_xor(sw, o, 32);
    swx1 += __shfl_xor(swx1, o, 32);
    swy1 += __shfl_xor(swy1, o, 32);
    swx2 += __shfl_xor(swx2, o, 32);
    swy2 += __shfl_xor(swy2, o, 32);
  }
  if (lane == 0) {
    atomicAdd(&sums[b * 8 + 0], sw);
    atomicAdd(&sums[b * 8 + 1], swx1);
    atomicAdd(&sums[b * 8 + 2], swy1);
    atomicAdd(&sums[b * 8 + 3], swx2);
    atomicAdd(&sums[b * 8 + 4], swy2);
  }
}

// ---------------------------------------------------------------------------
// Pass B: weighted mean radius (needs centroids from pass A). L2-resident.
// ---------------------------------------------------------------------------
__global__ __launch_bounds__(256) void radius_kernel(
    const float* __restrict__ pts1, const float* __restrict__ pts2,
    const float* __restrict__ wgt, float* __restrict__ sums,
    int B, int N, int chunks)
{
  const int wg = blockIdx.x;
  const int b = wg / chunks, c = wg % chunks;
  const int tid = threadIdx.x, lane = tid & 31;

  const float inv = 1.0f / sums[b * 8 + 0];
  const float cx1 = sums[b * 8 + 1] * inv, cy1 = sums[b * 8 + 2] * inv;
  const float cx2 = sums[b * 8 + 3] * inv, cy2 = sums[b * 8 + 4] * inv;

  const float* p1b = pts1 + (size_t)b * N * 3;
  const float* p2b = pts2 + (size_t)b * N * 3;
  const float* wb  = wgt  + (size_t)b * N;

  float a1 = 0.f, a2 = 0.f;
  for (int k = 0; k < CHUNK_PTS / 256; ++k) {
    int n = c * CHUNK_PTS + k * 256 + tid;
    if (n < N) {
      float wv  = wb[n];
      float dx1 = p1b[3 * n + 0] - cx1, dy1 = p1b[3 * n + 1] - cy1;
      float dx2 = p2b[3 * n + 0] - cx2, dy2 = p2b[3 * n + 1] - cy2;
      a1 += wv * sqrtf(dx1 * dx1 + dy1 * dy1);
      a2 += wv * sqrtf(dx2 * dx2 + dy2 * dy2);
    }
  }
  for (int o = 16; o > 0; o >>= 1) {
    a1 += __shfl_xor(a1, o, 32);
    a2 += __shfl_xor(a2, o, 32);
  }
  if (lane == 0) {
    atomicAdd(&sums[b * 8 + 5], a1);
    atomicAdd(&sums[b * 8 + 6], a2);
  }
}

// ---------------------------------------------------------------------------
// Cyclic Jacobi eigensolver for small symmetric matrices (columns of V = evecs)
// ---------------------------------------------------------------------------
__device__ void jacobi_sym(float* A, float* V, int n, int sweeps) {
  for (int i = 0; i < n; ++i)
    for (int j = 0; j < n; ++j) V[i * n + j] = (i == j) ? 1.0f : 0.0f;
  for (int s = 0; s < sweeps; ++s) {
    for (int p = 0; p < n - 1; ++p) {
      for (int q = p + 1; q < n; ++q) {
        float apq = A[p * n + q];
        if (fabsf(apq) < 1e-20f) continue;
        float app = A[p * n + p], aqq = A[q * n + q];
        float theta = 0.5f * (aqq - app) / apq;
        float t = ((theta >= 0.f) ? 1.0f : -1.0f) /
                  (fabsf(theta) + sqrtf(theta * theta + 1.0f));
        float cth = 1.0f / sqrtf(t * t + 1.0f);
        float sth = t * cth;
        for (int k = 0; k < n; ++k) {   // columns p,q
          float akp = A[k * n + p], akq = A[k * n + q];
          A[k * n + p] = cth * akp - sth * akq;
          A[k * n + q] = sth * akp + cth * akq;
        }
        for (int k = 0; k < n; ++k) {   // rows p,q
          float apk = A[p * n + k], aqk = A[q * n + k];
          A[p * n + k] = cth * apk - sth * aqk;
          A[q * n + k] = sth * apk + cth * aqk;
        }
        for (int k = 0; k < n; ++k) {   // eigenvector columns
          float vkp = V[k * n + p], vkq = V[k * n + q];
          V[k * n + p] = cth * vkp - sth * vkq;
          V[k * n + q] = sth * vkp + cth * vkq;
        }
      }
    }
  }
}

// ---------------------------------------------------------------------------
// Finalize: one thread per batch (128 threads total; work is tiny).
// ---------------------------------------------------------------------------
__global__ void finalize_kernel(const float* __restrict__ sums,
                                const float* __restrict__ M0g,
                                float* __restrict__ out, int B) {
  int b = blockIdx.x * blockDim.x + threadIdx.x;
  if (b >= B) return;
  const float* s = sums + b * 8;
  float inv = 1.0f / s[0];
  float cx1 = s[1] * inv, cy1 = s[2] * inv;
  float cx2 = s[3] * inv, cy2 = s[4] * inv;
  float sc1 = 1.4142f / (s[5] * inv);
  float sc2 = 1.4142f / (s[6] * inv);

  float T1[3][3] = {{sc1, 0.f, -cx1 * sc1}, {0.f, sc1, -cy1 * sc1}, {0.f, 0.f, 1.f}};
  float T2[3][3] = {{sc2, 0.f, -cx2 * sc2}, {0.f, sc2, -cy2 * sc2}, {0.f, 0.f, 1.f}};

  // K = kron(T1, T2): normalized design vec = K * raw kron vec
  float K[9][9];
  for (int i = 0; i < 3; ++i)
    for (int j = 0; j < 3; ++j)
      for (int k = 0; k < 3; ++k)
        for (int l = 0; l < 3; ++l)
          K[3 * i + j][3 * k + l] = T1[i][k] * T2[j][l];

  float Mw[9][9];
  for (int r = 0; r < 9; ++r)
    for (int cidx = 0; cidx < 9; ++cidx)
      Mw[r][cidx] = M0g[(size_t)b * 256 + r * 16 + cidx];

  // M = K * M0 * K^T + AUG*I
  float tmp[9][9], M[81];
  for (int i = 0; i < 9; ++i)
    for (int j = 0; j < 9; ++j) {
      float a = 0.f;
      for (int k = 0; k < 9; ++k) a += K[i][k] * Mw[k][j];
      tmp[i][j] = a;
    }
  for (int i = 0; i < 9; ++i)
    for (int j = 0; j < 9; ++j) {
      float a = 0.f;
      for (int k = 0; k < 9; ++k) a += tmp[i][k] * K[j][k];
      M[i * 9 + j] = a + ((i == j) ? AUG : 0.f);
    }

  // smallest eigenvector of 9x9
  float V9[81];
  jacobi_sym(M, V9, 9, 15);
  int idx = 0;
  for (int d = 1; d < 9; ++d)
    if (M[d * 9 + d] < M[idx * 9 + idx]) idx = d;
  float F[3][3];
  for (int i = 0; i < 3; ++i)
    for (int j = 0; j < 3; ++j) F[i][j] = V9[(3 * i + j) * 9 + idx];

  // rank-2 projection: Fp = (I - u0 u0^T) F, u0 = min-evec of F F^T + AUG I
  float G[9];
  for (int i = 0; i < 3; ++i)
    for (int j = 0; j < 3; ++j) {
      float a = 0.f;
      for (int k = 0; k < 3; ++k) a += F[i][k] * F[j][k];
      G[i * 3 + j] = a + ((i == j) ? AUG : 0.f);
    }
  float U3[9];
  jacobi_sym(G, U3, 3, 8);
  int i3 = 0;
  for (int d = 1; d < 3; ++d)
    if (G[d * 3 + d] < G[i3 * 3 + i3]) i3 = d;
  float u0[3] = {U3[0 * 3 + i3], U3[1 * 3 + i3], U3[2 * 3 + i3]};
  float rrow[3];
  for (int j = 0; j < 3; ++j)
    rrow[j] = u0[0] * F[0][j] + u0[1] * F[1][j] + u0[2] * F[2][j];
  float Fp[3][3];
  for (int i = 0; i < 3; ++i)
    for (int j = 0; j < 3; ++j) Fp[i][j] = F[i][j] - u0[i] * rrow[j];

  // out = T1^T * Fp * T2
  float A1[3][3];
  for (int i = 0; i < 3; ++i)
    for (int j = 0; j < 3; ++j) {
      float a = 0.f;
      for (int k = 0; k < 3; ++k) a += T1[k][i] * Fp[k][j];
      A1[i][j] = a;
    }
  for (int i = 0; i < 3; ++i)
    for (int j = 0; j < 3; ++j) {
      float a = 0.f;
      for (int k = 0; k < 3; ++k) a += A1[i][k] * T2[k][j];
      out[(size_t)b * 9 + i * 3 + j] = a;
    }
}

// ---------------------------------------------------------------------------
extern "C" void kernel_launch(void* const* d_in, const int* in_sizes, int n_in,
                              void* d_out, int out_size, void* d_ws, size_t ws_size,
                              hipStream_t stream) {
  const float* pts1 = (const float*)d_in[0];
  const float* pts2 = (const float*)d_in[1];
  const float* wgt  = (const float*)d_in[2];
  float* out = (float*)d_out;

  const int B = out_size / 9;              // output [B,3,3]
  const int N = in_sizes[2] / B;           // weights [B,1,N]

  // ws layout (floats): [B*8] centroid/radius sums | [B*256] M0 accumulators
  float* sums = (float*)d_ws;
  float* M0   = sums + (size_t)B * 8;
  const int zn = B * (8 + 256);
  const int chunks = (N + CHUNK_PTS - 1) / CHUNK_PTS;

  hipLaunchKernelGGL(zero_ws_kernel, dim3((zn + 255) / 256), dim3(256), 0, stream,
                     (float*)d_ws, zn);
  hipLaunchKernelGGL(moments_kernel, dim3(B * chunks), dim3(256), 0, stream,
                     pts1, pts2, wgt, sums, M0, B, N, chunks);
  hipLaunchKernelGGL(radius_kernel, dim3(B * chunks), dim3(256), 0, stream,
                     pts1, pts2, wgt, sums, B, N, chunks);
  hipLaunchKernelGGL(finalize_kernel, dim3((B + 127) / 128), dim3(128), 0, stream,
                     sums, M0, out, B);
}